// GNNActor_56375740727650
// MI455X (gfx1250) — compile-verified
//
#include <hip/hip_runtime.h>
#include <hip/hip_bf16.h>

typedef float v2f __attribute__((ext_vector_type(2)));
typedef float v8f __attribute__((ext_vector_type(8)));

#define NN 100000
#define NE 3200000
#define WAVES 8   // 256 threads = 8 wave32

// ---------------------------------------------------------------------------
// LDS layout for the edge-MLP kernel.
// w1 is staged padded to 68 rows (rows 66,67 zeroed) with row stride 33.
// feat[wave] holds the 16x68 fp32 A-tile; after layer 1 the same region is
// reused (stride 34) for the 16x32 hidden activations (bank-conflict-free
// for the A-fragment reads: 34*e mod 64 is distinct for e=0..15).
// ---------------------------------------------------------------------------
struct __align__(16) EdgeSmem {
  float w1[68 * 33];
  float w2[32 * 33];
  float bb1[32];
  float bb2[32];
  float feat[WAVES][16 * 68];
  int   srcid[WAVES][16];
};

__device__ __forceinline__ float softplus_f(float z) {
  // numerically stable log(1+exp(z)) as in jax.nn.softplus
  return fmaxf(z, 0.f) + log1pf(__expf(-fabsf(z)));
}

__global__ __launch_bounds__(256) void edge_mlp_scatter_kernel(
    const float* __restrict__ x, const int* __restrict__ ei,
    const float* __restrict__ eattr,
    const float* __restrict__ W1, const float* __restrict__ B1,
    const float* __restrict__ W2, const float* __restrict__ B2,
    float* __restrict__ agg) {
  __shared__ EdgeSmem sm;
  const int tid = threadIdx.x;

  // ---- stage weights into LDS (once per block) ----
  for (int i = tid; i < 66 * 32; i += 256) sm.w1[(i >> 5) * 33 + (i & 31)] = W1[i];
  for (int i = tid; i < 2 * 32; i += 256) sm.w1[(66 + (i >> 5)) * 33 + (i & 31)] = 0.f;
  for (int i = tid; i < 32 * 32; i += 256) sm.w2[(i >> 5) * 33 + (i & 31)] = W2[i];
  if (tid < 32) sm.bb1[tid] = B1[tid];
  else if (tid < 64) sm.bb2[tid - 32] = B2[tid - 32];
  __syncthreads();

  const int wave = tid >> 5;
  const int lane = tid & 31;
  const int mrow = lane & 15;   // C/D: N column; A: M row
  const int hi   = lane >> 4;   // lane-group select (K split for A/B, M split for C/D)
  const int nTiles = NE / 16;   // 200000, exact
  float* const fbase = sm.feat[wave];

  // ---- hoist ALL loop-invariant B fragments (W1, W2) into registers ----
  v2f w1f[17][2];
  #pragma unroll
  for (int ks = 0; ks < 17; ks++) {
    const float* wp = sm.w1 + (ks * 4 + 2 * hi) * 33;
    w1f[ks][0].x = wp[mrow];      w1f[ks][0].y = wp[33 + mrow];
    w1f[ks][1].x = wp[16 + mrow]; w1f[ks][1].y = wp[33 + 16 + mrow];
  }
  v2f w2f[8][2];
  #pragma unroll
  for (int ks = 0; ks < 8; ks++) {
    const float* wp = sm.w2 + (ks * 4 + 2 * hi) * 33;
    w2f[ks][0].x = wp[mrow];      w2f[ks][0].y = wp[33 + mrow];
    w2f[ks][1].x = wp[16 + mrow]; w2f[ks][1].y = wp[33 + 16 + mrow];
  }
  const float t0 = sm.bb1[mrow], t1 = sm.bb1[16 + mrow];   // layer-1 bias (per N col)
  const float o0 = sm.bb2[mrow], o1 = sm.bb2[16 + mrow];   // layer-2 bias
  const float* const fA = fbase + mrow * 68 + 2 * hi;      // layer-1 A fragment base
  const float* const hA = fbase + mrow * 34 + 2 * hi;      // layer-2 A fragment base

  for (int tile = blockIdx.x * WAVES + wave; tile < nTiles; tile += gridDim.x * WAVES) {
    const int e0 = tile * 16;

    // ---- gather: build 16x68 concat(x_i, x_j, edge_attr, 0,0) in LDS ----
    {
      const int e = lane >> 1, h = lane & 1;  // 2 lanes per edge
      const int sn = ei[e0 + e];
      const int dn = ei[NE + e0 + e];
      if (!h) sm.srcid[wave][e] = sn;
      const float2* xi = (const float2*)x + (size_t)sn * 11;
      const float2* xj = (const float2*)x + (size_t)dn * 11;
      const float2* ev = (const float2*)eattr + (size_t)(e0 + e) * 11;
      float2* frow = (float2*)(fbase + e * 68);
      for (int j = h; j < 33; j += 2) {  // 33 float2 = 66 floats per edge
        float2 v;
        if (j < 11)      v = xi[j];
        else if (j < 22) v = xj[j - 11];
        else             v = ev[j - 22];
        frow[j] = v;
      }
      if (!h) frow[33] = make_float2(0.f, 0.f);  // K-pad 66..67
    }
    asm volatile("s_wait_dscnt 0" ::: "memory");  // wave-local LDS RAW fence

    // ---- layer 1: [16x68] @ W1[68x32] via v_wmma_f32_16x16x4_f32 ----
    // bias folded into accumulator init (bias depends only on N column)
    v8f c0 = {t0, t0, t0, t0, t0, t0, t0, t0};
    v8f c1 = {t1, t1, t1, t1, t1, t1, t1, t1};
    #pragma unroll
    for (int ks = 0; ks < 17; ks++) {
      v2f a = *(const v2f*)(fA + ks * 4);  // A: K=4ks+2hi, 4ks+2hi+1
      c0 = __builtin_amdgcn_wmma_f32_16x16x4_f32(false, a, false, w1f[ks][0], (short)0, c0, false, false);
      c1 = __builtin_amdgcn_wmma_f32_16x16x4_f32(false, a, false, w1f[ks][1], (short)0, c1, false, false);
    }

    // ---- ReLU, then transpose C-layout -> A-layout through LDS ----
    #pragma unroll
    for (int r = 0; r < 8; r++) {
      c0[r] = fmaxf(c0[r], 0.f);
      c1[r] = fmaxf(c1[r], 0.f);
    }
    #pragma unroll
    for (int r = 0; r < 8; r++) {
      const int m = r + 8 * hi;             // edge row held by this VGPR
      fbase[m * 34 + mrow]      = c0[r];    // reuse feat region, stride 34
      fbase[m * 34 + 16 + mrow] = c1[r];
    }
    asm volatile("s_wait_dscnt 0" ::: "memory");

    // ---- layer 2: [16x32] @ W2[32x32] ----
    v8f d0 = {o0, o0, o0, o0, o0, o0, o0, o0};
    v8f d1 = {o1, o1, o1, o1, o1, o1, o1, o1};
    #pragma unroll
    for (int ks = 0; ks < 8; ks++) {
      v2f a = *(const v2f*)(hA + ks * 4);
      d0 = __builtin_amdgcn_wmma_f32_16x16x4_f32(false, a, false, w2f[ks][0], (short)0, d0, false, false);
      d1 = __builtin_amdgcn_wmma_f32_16x16x4_f32(false, a, false, w2f[ks][1], (short)0, d1, false, false);
    }

    // ---- fused segment_sum: atomic scatter-add into agg[src] ----
    #pragma unroll
    for (int r = 0; r < 8; r++) {
      const int m = r + 8 * hi;
      const int node = sm.srcid[wave][m];
      atomicAdd(agg + (size_t)node * 32 + mrow,      d0[r]);
      atomicAdd(agg + (size_t)node * 32 + 16 + mrow, d1[r]);
    }
  }
}

// ---------------------------------------------------------------------------
__global__ __launch_bounds__(256) void zero_f4_kernel(float4* __restrict__ p, int n4) {
  int i = blockIdx.x * blockDim.x + threadIdx.x;
  const int stride = gridDim.x * blockDim.x;
  const float4 z = make_float4(0.f, 0.f, 0.f, 0.f);
  for (; i < n4; i += stride) p[i] = z;
}

// ---------------------------------------------------------------------------
__global__ __launch_bounds__(256) void heads_kernel(
    const float* __restrict__ x, const float* __restrict__ agg,
    const float* __restrict__ Wmu, const float* __restrict__ bmu,
    const float* __restrict__ Wsig, const float* __restrict__ bsig,
    const float* __restrict__ Wconc, const float* __restrict__ bconc,
    float* __restrict__ out) {
  __shared__ float w[3][54];
  __shared__ float bb[3];
  const int tid = threadIdx.x;
  if (tid < 54) {
    w[0][tid] = Wmu[tid];
    w[1][tid] = Wsig[tid];
    w[2][tid] = Wconc[tid];
  }
  if (tid == 0) { bb[0] = bmu[0]; bb[1] = bsig[0]; bb[2] = bconc[0]; }
  __syncthreads();

  const int n = blockIdx.x * blockDim.x + tid;
  if (n >= NN) return;
  float a0 = bb[0], a1 = bb[1], a2 = bb[2];
  const float* xr = x + (size_t)n * 22;
  #pragma unroll
  for (int c = 0; c < 22; c++) {
    const float v = xr[c];
    a0 = fmaf(v, w[0][c], a0); a1 = fmaf(v, w[1][c], a1); a2 = fmaf(v, w[2][c], a2);
  }
  const float* ar = agg + (size_t)n * 32;
  #pragma unroll
  for (int c = 0; c < 32; c++) {
    const float v = ar[c];
    a0 = fmaf(v, w[0][22 + c], a0); a1 = fmaf(v, w[1][22 + c], a1); a2 = fmaf(v, w[2][22 + c], a2);
  }
  out[n]          = softplus_f(a0);  // mu
  out[NN + n]     = softplus_f(a1);  // sigma
  out[2 * NN + n] = softplus_f(a2);  // alpha
}

// ---------------------------------------------------------------------------
extern "C" void kernel_launch(void* const* d_in, const int* in_sizes, int n_in,
                              void* d_out, int out_size, void* d_ws, size_t ws_size,
                              hipStream_t stream) {
  (void)in_sizes; (void)n_in; (void)out_size; (void)ws_size;
  const float* x     = (const float*)d_in[0];
  const int*   ei    = (const int*)d_in[1];    // edge_index flat [2][E]
  const float* eattr = (const float*)d_in[2];
  const float* W1    = (const float*)d_in[3];
  const float* B1    = (const float*)d_in[4];
  const float* W2    = (const float*)d_in[5];
  const float* B2    = (const float*)d_in[6];
  const float* Wmu   = (const float*)d_in[7];
  const float* bmu   = (const float*)d_in[8];
  const float* Wsig  = (const float*)d_in[9];
  const float* bsig  = (const float*)d_in[10];
  const float* Wconc = (const float*)d_in[11];
  const float* bconc = (const float*)d_in[12];
  float* out = (float*)d_out;
  float* agg = (float*)d_ws;  // NN*32 floats = 12.8 MB scratch

  // 1) zero the aggregation buffer (atomics accumulate into it)
  zero_f4_kernel<<<1024, 256, 0, stream>>>((float4*)agg, NN * 32 / 4);
  // 2) edge MLP (fp32 WMMA) + fused atomic segment-sum
  edge_mlp_scatter_kernel<<<2000, 256, 0, stream>>>(x, ei, eattr, W1, B1, W2, B2, agg);
  // 3) per-node heads + softplus
  heads_kernel<<<(NN + 255) / 256, 256, 0, stream>>>(
      x, agg, Wmu, bmu, Wsig, bsig, Wconc, bconc, out);
}